// DigitCaps_79912161509769
// MI455X (gfx1250) — compile-verified
//
#include <hip/hip_runtime.h>
#include <math.h>

typedef __attribute__((ext_vector_type(2))) float v2f;
typedef __attribute__((ext_vector_type(8))) float v8f;

#define SI 8192
#define IN_CAPS 50
#define IN_SIZE 9
#define OUT_CAPS 29
#define OUT_SIZE 8
#define K_DIM 450   // IN_CAPS*IN_SIZE
#define KP 464      // K padded to multiple of 16 (also M of P-GEMM)
#define N_DIM 232   // OUT_CAPS*OUT_SIZE
#define NP 240      // N padded to multiple of 16
#define PSPLIT 8    // split-K factor for the P GEMM (8192 -> 8 x 1024)

// workspace layout (float offsets)
#define OFF_U   0
#define OFF_MC  (SI * KP)
#define OFF_V   (OFF_MC + NP * KP)
#define OFF_P   (OFF_V + SI * NP)                 // PSPLIT partial P matrices
#define OFF_B   (OFF_P + PSPLIT * KP * NP)
#define OFF_C   (OFF_B + IN_CAPS * OUT_CAPS)

// ---------------------------------------------------------------------------
// Conv 10x10 stride 5 on 20x20 -> u[s, i*9+p], zero-padded to KP columns.
// ---------------------------------------------------------------------------
__global__ void conv_u_kernel(const float* __restrict__ x,
                              const float* __restrict__ cw,
                              const float* __restrict__ cb,
                              float* __restrict__ U) {
  int gid = blockIdx.x * blockDim.x + threadIdx.x;   // SI*KP threads, exact grid
  int s = gid / KP;
  int k = gid - s * KP;
  float acc = 0.f;
  if (k < K_DIM) {
    int i = k / IN_SIZE;
    int p = k - i * IN_SIZE;
    int py = p / 3, px = p - py * 3;
    const float* xr = x + s * 400 + (py * 5) * 20 + px * 5;
    const float* wr = cw + i * 100;
    acc = cb[i];
#pragma unroll
    for (int ky = 0; ky < 10; ++ky)
#pragma unroll
      for (int kx = 0; kx < 10; ++kx)
        acc = fmaf(xr[ky * 20 + kx], wr[ky * 10 + kx], acc);
  }
  U[gid] = acc;
}

// ---------------------------------------------------------------------------
// c[i,:] = log_softmax(b[i,:])
// ---------------------------------------------------------------------------
__global__ void coeff_kernel(const float* __restrict__ B, float* __restrict__ C) {
  int i = threadIdx.x;
  if (i >= IN_CAPS) return;
  float m = -3.402823466e38f;
  for (int j = 0; j < OUT_CAPS; ++j) m = fmaxf(m, B[i * OUT_CAPS + j]);
  float ssum = 0.f;
  for (int j = 0; j < OUT_CAPS; ++j) ssum += expf(B[i * OUT_CAPS + j] - m);
  float lse = m + logf(ssum);
  for (int j = 0; j < OUT_CAPS; ++j) C[i * OUT_CAPS + j] = B[i * OUT_CAPS + j] - lse;
}

// ---------------------------------------------------------------------------
// Mc[ja, k] = c[i,j] * W[i,j,a,b]  with ja=j*8+a, k=i*9+b; zero padded.
// Row-major (NP rows, KP cols).
// ---------------------------------------------------------------------------
__global__ void mc_kernel(const float* __restrict__ C, const float* __restrict__ W,
                          float* __restrict__ MC) {
  int gid = blockIdx.x * blockDim.x + threadIdx.x;   // NP*KP threads, exact grid
  int ja = gid / KP;
  int k = gid - ja * KP;
  float val = 0.f;
  if (ja < N_DIM && k < K_DIM) {
    int j = ja >> 3, a = ja & 7;
    int i = k / IN_SIZE, bb = k - i * IN_SIZE;
    val = C[i * OUT_CAPS + j] * W[((i * OUT_CAPS + j) * OUT_SIZE + a) * IN_SIZE + bb];
  }
  MC[gid] = val;
}

// ---------------------------------------------------------------------------
// S = U(8192 x KP) * Mc^T  -> per-wave 64x16 tile via 4x V_WMMA_F32_16X16X4_F32
// Epilogue via LDS: mode 0 -> squash into V (8192 x NP, pad cols are 0 naturally)
//                   mode 1 -> out[s,j] = ||S[s,j,:]||
// ---------------------------------------------------------------------------
__global__ __launch_bounds__(256) void s_gemm_kernel(
    const float* __restrict__ U, const float* __restrict__ MC,
    float* __restrict__ V, float* __restrict__ out, int mode) {
  __shared__ float lds[8 * 64 * 16];   // 32 KB: 64x16 tile per wave
  int lane = threadIdx.x & 31;
  int wv = threadIdx.x >> 5;
  int gt = blockIdx.x * 8 + wv;        // 0..1919, exact
  int nt = gt % 15;                    // N tile
  int mt = gt / 15;                    // M tile group (64 rows)
  int half = lane >> 4;
  int l16 = lane & 15;
  int mrow0 = mt * 64;
  int kb = half * 2;

  v8f acc0 = {}, acc1 = {}, acc2 = {}, acc3 = {};
  const float* Arow = U + (mrow0 + l16) * KP;
  const float* Brow = MC + (nt * 16 + l16) * KP;

  for (int k0 = 0; k0 < KP; k0 += 4) {
    v2f b  = *(const v2f*)(Brow + k0 + kb);
    v2f a0 = *(const v2f*)(Arow + k0 + kb);
    v2f a1 = *(const v2f*)(Arow + 16 * KP + k0 + kb);
    v2f a2 = *(const v2f*)(Arow + 32 * KP + k0 + kb);
    v2f a3 = *(const v2f*)(Arow + 48 * KP + k0 + kb);
    acc0 = __builtin_amdgcn_wmma_f32_16x16x4_f32(false, a0, false, b, (short)0, acc0, false, false);
    acc1 = __builtin_amdgcn_wmma_f32_16x16x4_f32(false, a1, false, b, (short)0, acc1, false, false);
    acc2 = __builtin_amdgcn_wmma_f32_16x16x4_f32(false, a2, false, b, (short)0, acc2, false, false);
    acc3 = __builtin_amdgcn_wmma_f32_16x16x4_f32(false, a3, false, b, (short)0, acc3, false, false);
  }

  float* L = lds + wv * 1024;
#pragma unroll
  for (int r = 0; r < 8; ++r) {
    int rr = r + half * 8;             // C layout: m = r + (lane>>4)*8, n = lane&15
    L[( 0 + rr) * 16 + l16] = acc0[r];
    L[(16 + rr) * 16 + l16] = acc1[r];
    L[(32 + rr) * 16 + l16] = acc2[r];
    L[(48 + rr) * 16 + l16] = acc3[r];
  }
  __syncthreads();

#pragma unroll
  for (int q = 0; q < 4; ++q) {        // 128 (row, j-half) pairs per wave
    int p = q * 32 + lane;
    int row = p >> 1;
    int jj = p & 1;
    const float* src = L + row * 16 + jj * 8;
    float sv[8];
    float sq = 0.f;
#pragma unroll
    for (int a = 0; a < 8; ++a) { sv[a] = src[a]; sq += sv[a] * sv[a]; }
    int row_g = mrow0 + row;
    if (mode == 0) {
      float scale = (sq / (1.f + sq)) / (sqrtf(sq) + 1e-12f);
      float4 lo = { sv[0] * scale, sv[1] * scale, sv[2] * scale, sv[3] * scale };
      float4 hi = { sv[4] * scale, sv[5] * scale, sv[6] * scale, sv[7] * scale };
      float4* dst = (float4*)(V + row_g * NP + nt * 16 + jj * 8);
      dst[0] = lo;
      dst[1] = hi;
    } else {
      int j = nt * 2 + jj;
      if (j < OUT_CAPS) out[row_g * OUT_CAPS + j] = sqrtf(sq);
    }
  }
}

// ---------------------------------------------------------------------------
// Split-K partials: Pp[sp][k, ja] = sum_{s in chunk sp} U[s,k] * V[s,ja]
// 29*15*PSPLIT = 3480 wave-tiles (435 blocks x 8 waves, exact grid).
// Two independent accumulator chains break the WMMA RAW dependency.
// ---------------------------------------------------------------------------
__global__ __launch_bounds__(256) void p_gemm_kernel(
    const float* __restrict__ U, const float* __restrict__ V, float* __restrict__ Pp) {
  int lane = threadIdx.x & 31;
  int wv = threadIdx.x >> 5;
  int gt = blockIdx.x * 8 + wv;          // 0..3479, exact
  int tile = gt % ((KP / 16) * (NP / 16));
  int sp = gt / ((KP / 16) * (NP / 16)); // split index 0..PSPLIT-1
  int nt = tile % (NP / 16);
  int mt = tile / (NP / 16);
  int l16 = lane & 15;
  int kk = (lane >> 4) * 2;
  int m = mt * 16 + l16;
  int n = nt * 16 + l16;
  int sBeg = sp * (SI / PSPLIT);
  int sEnd = sBeg + (SI / PSPLIT);

  v8f accA = {}, accB = {};
  for (int s0 = sBeg; s0 < sEnd; s0 += 8) {
    v2f a0, b0, a1, b1;
    a0.x = U[(s0 + kk    ) * KP + m];
    a0.y = U[(s0 + kk + 1) * KP + m];
    b0.x = V[(s0 + kk    ) * NP + n];
    b0.y = V[(s0 + kk + 1) * NP + n];
    a1.x = U[(s0 + 4 + kk    ) * KP + m];
    a1.y = U[(s0 + 4 + kk + 1) * KP + m];
    b1.x = V[(s0 + 4 + kk    ) * NP + n];
    b1.y = V[(s0 + 4 + kk + 1) * NP + n];
    accA = __builtin_amdgcn_wmma_f32_16x16x4_f32(false, a0, false, b0, (short)0, accA, false, false);
    accB = __builtin_amdgcn_wmma_f32_16x16x4_f32(false, a1, false, b1, (short)0, accB, false, false);
  }
  v8f acc = accA + accB;
  float* P = Pp + sp * (KP * NP);
#pragma unroll
  for (int r = 0; r < 8; ++r) {
    int mr = mt * 16 + r + (lane >> 4) * 8;
    P[mr * NP + nt * 16 + l16] = acc[r];
  }
}

// ---------------------------------------------------------------------------
// b[i,j] += (1/SI) * sum_{a,bb} W[i,j,a,bb] * sum_sp Pp[sp][i*9+bb, j*8+a]
// Fixed summation order -> deterministic.
// ---------------------------------------------------------------------------
__global__ void b_update_kernel(const float* __restrict__ W, const float* __restrict__ Pp,
                                float* __restrict__ B) {
  int idx = blockIdx.x * blockDim.x + threadIdx.x;
  if (idx >= IN_CAPS * OUT_CAPS) return;
  int i = idx / OUT_CAPS, j = idx - i * OUT_CAPS;
  float acc = 0.f;
#pragma unroll
  for (int a = 0; a < 8; ++a)
#pragma unroll
    for (int bb = 0; bb < 9; ++bb) {
      int off = (i * IN_SIZE + bb) * NP + j * OUT_SIZE + a;
      float ps = 0.f;
#pragma unroll
      for (int sp = 0; sp < PSPLIT; ++sp) ps += Pp[sp * (KP * NP) + off];
      acc = fmaf(W[((i * OUT_CAPS + j) * OUT_SIZE + a) * IN_SIZE + bb], ps, acc);
    }
  B[idx] += acc * (1.f / (float)SI);
}

// ---------------------------------------------------------------------------
extern "C" void kernel_launch(void* const* d_in, const int* in_sizes, int n_in,
                              void* d_out, int out_size, void* d_ws, size_t ws_size,
                              hipStream_t stream) {
  const float* x  = (const float*)d_in[0];
  const float* W  = (const float*)d_in[1];
  const float* cw = (const float*)d_in[2];
  const float* cb = (const float*)d_in[3];
  float* out = (float*)d_out;
  float* ws = (float*)d_ws;
  float* U  = ws + OFF_U;
  float* MC = ws + OFF_MC;
  float* V  = ws + OFF_V;
  float* Pp = ws + OFF_P;
  float* B  = ws + OFF_B;
  float* C  = ws + OFF_C;

  (void)in_sizes; (void)n_in; (void)out_size; (void)ws_size;

  hipMemsetAsync(B, 0, IN_CAPS * OUT_CAPS * sizeof(float), stream);
  conv_u_kernel<<<(SI * KP) / 256, 256, 0, stream>>>(x, cw, cb, U);

  for (int it = 0; it < 3; ++it) {
    coeff_kernel<<<1, 64, 0, stream>>>(B, C);
    mc_kernel<<<(NP * KP) / 256, 256, 0, stream>>>(C, W, MC);
    if (it < 2) {
      s_gemm_kernel<<<240, 256, 0, stream>>>(U, MC, V, out, 0);
      p_gemm_kernel<<<((KP / 16) * (NP / 16) * PSPLIT) / 8, 256, 0, stream>>>(U, V, Pp);
      b_update_kernel<<<6, 256, 0, stream>>>(W, Pp, B);
    } else {
      s_gemm_kernel<<<240, 256, 0, stream>>>(U, MC, V, out, 1);
    }
  }
}